// PositionFeaturizer_36077725286973
// MI455X (gfx1250) — compile-verified
//
#include <hip/hip_runtime.h>
#include <math.h>

// Problem constants (match reference)
#define NROWS  32768
#define EDIM   512
#define HHEADS 8
#define DDIM   64
#define NNZE   524288

typedef __attribute__((ext_vector_type(2))) float v2f;
typedef __attribute__((ext_vector_type(8))) float v8f;

// ---------------------------------------------------------------------------
// Kernel 1: Q/K projections with f32 WMMA (V_WMMA_F32_16X16X4_F32).
//   q = (x @ q_w^T + q_b) * (1/sqrt(D)) ; k = x @ k_w^T + k_b
// One wave computes a 16(M) x 32(N) strip (two 16x16 tiles, A-fragment reuse).
// A-fragment (16x4 f32): lane<16 -> M=lane, K=2*half+{0,1}; symmetric for B.
// C/D layout: VGPR r, lane<16 -> (M=r, N=lane); lane>=16 -> (M=r+8, N=lane-16).
// ---------------------------------------------------------------------------
__global__ __launch_bounds__(256) void qk_gemm_wmma(
    const float* __restrict__ x,
    const float* __restrict__ q_w, const float* __restrict__ q_b, float* __restrict__ q_out,
    const float* __restrict__ k_w, const float* __restrict__ k_b, float* __restrict__ k_out)
{
  const float* w; const float* b; float* o; float scale;
  if (blockIdx.y == 0) { w = q_w; b = q_b; o = q_out; scale = 0.125f; }   // 1/sqrt(64)
  else                 { w = k_w; b = k_b; o = k_out; scale = 1.0f;  }

  const int wave = (blockIdx.x * blockDim.x + threadIdx.x) >> 5;
  const int lane = threadIdx.x & 31;
  const int mt = wave >> 4;            // 2048 M-tiles of 16 rows
  const int ns = wave & 15;            // 16 N-strips of 32 cols
  const int m0 = mt * 16, n0 = ns * 32;
  const int lr = lane & 15, half = lane >> 4;

  const float* arow  = x + (size_t)(m0 + lr) * EDIM + 2 * half;
  const float* brow0 = w + (size_t)(n0 + lr) * EDIM + 2 * half;
  const float* brow1 = brow0 + (size_t)16 * EDIM;

  v8f acc0 = {}; v8f acc1 = {};
  for (int k0 = 0; k0 < EDIM; k0 += 4) {
    v2f a  = *(const v2f*)(arow  + k0);
    v2f b0 = *(const v2f*)(brow0 + k0);
    v2f b1 = *(const v2f*)(brow1 + k0);
    acc0 = __builtin_amdgcn_wmma_f32_16x16x4_f32(false, a, false, b0, (short)0, acc0, false, false);
    acc1 = __builtin_amdgcn_wmma_f32_16x16x4_f32(false, a, false, b1, (short)0, acc1, false, false);
  }

  const float bias0 = b[n0 + lr]      * scale;
  const float bias1 = b[n0 + 16 + lr] * scale;
  #pragma unroll
  for (int r = 0; r < 8; ++r) {
    const int m = m0 + r + 8 * half;
    const int n = n0 + lr;
    o[(size_t)m * EDIM + n]      = acc0[r] * scale + bias0;
    o[(size_t)m * EDIM + n + 16] = acc1[r] * scale + bias1;
  }
}

// ---------------------------------------------------------------------------
// Kernel 2: per-edge logits[e,h] = dot(q[row[e],h,:], k[tocol[col[e]],h,:]) + bias[h,e]
// One wave per edge: 32 lanes = 2 heads x (16 lanes x float4 = 64 floats).
// row_index is sorted -> q-row loads hit L0; k rows (64MB) live in the 192MB L2.
// ---------------------------------------------------------------------------
__global__ __launch_bounds__(256) void edge_logits(
    const float* __restrict__ q, const float* __restrict__ kproj,
    const int* __restrict__ row_index, const int* __restrict__ col_index,
    const int* __restrict__ to_col, const float* __restrict__ att_bias,
    float* __restrict__ logits)
{
  const int e    = (blockIdx.x * blockDim.x + threadIdx.x) >> 5;
  const int lane = threadIdx.x & 31;
  if (e >= NNZE) return;

  const int r = row_index[e];
  const int c = to_col[col_index[e]];
  const float4* qr = (const float4*)(q     + (size_t)r * EDIM);
  const float4* kr = (const float4*)(kproj + (size_t)c * EDIM);
  const int lr = lane & 15, half = lane >> 4;

  #pragma unroll
  for (int g = 0; g < 4; ++g) {
    const int idx = g * 32 + half * 16 + lr;   // float4 index; head = 2g + half
    const float4 a = qr[idx];
    const float4 b = kr[idx];
    float s = a.x * b.x + a.y * b.y + a.z * b.z + a.w * b.w;
    s += __shfl_xor(s, 1, 32);
    s += __shfl_xor(s, 2, 32);
    s += __shfl_xor(s, 4, 32);
    s += __shfl_xor(s, 8, 32);
    const int h = 2 * g + half;
    if (lr == 0)
      logits[(size_t)e * HHEADS + h] = s + att_bias[(size_t)h * NNZE + e];
  }
}

// ---------------------------------------------------------------------------
// Kernel 3: CSR row pointers from the sorted row_index (no atomics, no scan).
// ---------------------------------------------------------------------------
__global__ void build_row_ptr(const int* __restrict__ row_index, int* __restrict__ row_ptr)
{
  const int e = blockIdx.x * blockDim.x + threadIdx.x;
  if (e > NNZE) return;
  const int cur  = (e < NNZE) ? row_index[e]     : NROWS;
  const int prev = (e > 0)    ? row_index[e - 1] : -1;
  for (int r = prev + 1; r <= cur; ++r) row_ptr[r] = e;
}

// ---------------------------------------------------------------------------
// Kernel 4: fused per-row segment softmax + distance weighting + segment sums
// + normalize + output. One wave per row; lane = head(0..7) + 8 * slot(0..3).
// att is never materialized: out = (sum p*w*cp)/z etc. with z = sum p.
// ---------------------------------------------------------------------------
__global__ __launch_bounds__(256) void row_softmax_out(
    const float* __restrict__ logits, const int* __restrict__ row_ptr,
    const int* __restrict__ col_index, const float* __restrict__ dist,
    const float* __restrict__ pos, const float* __restrict__ col_pos,
    float* __restrict__ out)
{
  const int row  = (blockIdx.x * blockDim.x + threadIdx.x) >> 5;
  const int lane = threadIdx.x & 31;
  if (row >= NROWS) return;

  const int s = row_ptr[row], t = row_ptr[row + 1];
  const int h = lane & 7, j = lane >> 3;

  if (s == t) {                       // empty row -> all zeros (matches reference)
    if (j == 0) {
      out[(size_t)row * 32 + h * 4 + 0] = 0.f;
      out[(size_t)row * 32 + h * 4 + 1] = 0.f;
      out[(size_t)row * 32 + h * 4 + 2] = 0.f;
      out[(size_t)row * 32 + h * 4 + 3] = 0.f;
    }
    return;
  }

  // Pass 1: per-(row,head) max
  float m = -INFINITY;
  for (int e = s + j; e < t; e += 4)
    m = fmaxf(m, logits[(size_t)e * HHEADS + h]);
  m = fmaxf(m, __shfl_xor(m, 8, 32));
  m = fmaxf(m, __shfl_xor(m, 16, 32));

  // Pass 2: z = sum p; weighted sums of inv-dist and inv-dist * col_pos
  float z = 0.f, aw = 0.f, ax = 0.f, ay = 0.f, az = 0.f;
  for (int e = s + j; e < t; e += 4) {
    const float p  = expf(logits[(size_t)e * HHEADS + h] - m);
    const float dd = dist[e];
    const float wgt = (dd == 0.f) ? 0.f : p / dd;
    const int c = col_index[e];
    const float cx = col_pos[(size_t)c * 3 + 0];
    const float cy = col_pos[(size_t)c * 3 + 1];
    const float cz = col_pos[(size_t)c * 3 + 2];
    z += p; aw += wgt; ax += wgt * cx; ay += wgt * cy; az += wgt * cz;
  }
  z  += __shfl_xor(z, 8, 32);  z  += __shfl_xor(z, 16, 32);
  aw += __shfl_xor(aw, 8, 32); aw += __shfl_xor(aw, 16, 32);
  ax += __shfl_xor(ax, 8, 32); ax += __shfl_xor(ax, 16, 32);
  ay += __shfl_xor(ay, 8, 32); ay += __shfl_xor(ay, 16, 32);
  az += __shfl_xor(az, 8, 32); az += __shfl_xor(az, 16, 32);

  if (j == 0) {
    const float inv_z = 1.f / z;
    const float avg = aw * inv_z;
    const float px = pos[(size_t)row * 3 + 0];
    const float py = pos[(size_t)row * 3 + 1];
    const float pz = pos[(size_t)row * 3 + 2];
    const float vx = ax * inv_z - avg * px;
    const float vy = ay * inv_z - avg * py;
    const float vz = az * inv_z - avg * pz;
    const float nrm = sqrtf(vx * vx + vy * vy + vz * vz);
    const float inv_n = 1.f / fmaxf(nrm, 1e-12f);
    out[(size_t)row * 32 + h * 4 + 0] = vx * inv_n;
    out[(size_t)row * 32 + h * 4 + 1] = vy * inv_n;
    out[(size_t)row * 32 + h * 4 + 2] = vz * inv_n;
    out[(size_t)row * 32 + h * 4 + 3] = avg;
  }
}

// ---------------------------------------------------------------------------
extern "C" void kernel_launch(void* const* d_in, const int* in_sizes, int n_in,
                              void* d_out, int out_size, void* d_ws, size_t ws_size,
                              hipStream_t stream) {
  const float* x         = (const float*)d_in[0];
  const int*   row_index = (const int*)  d_in[1];
  const int*   col_index = (const int*)  d_in[2];
  const int*   to_col    = (const int*)  d_in[3];
  const float* att_bias  = (const float*)d_in[4];
  const float* dist      = (const float*)d_in[5];
  const float* pos       = (const float*)d_in[6];
  const float* col_pos   = (const float*)d_in[7];
  const float* q_w       = (const float*)d_in[8];
  const float* q_b       = (const float*)d_in[9];
  const float* k_w       = (const float*)d_in[10];
  const float* k_b       = (const float*)d_in[11];
  float* out = (float*)d_out;

  // Workspace layout (~144.1 MB): q (64MB) | k (64MB) | logits (16MB) | row_ptr
  float* q_out  = (float*)d_ws;
  float* k_out  = q_out + (size_t)NROWS * EDIM;
  float* logits = k_out + (size_t)NROWS * EDIM;
  int*   row_ptr = (int*)(logits + (size_t)NNZE * HHEADS);

  // 1) Q & K projections: 2048 M-tiles x 16 N-strips = 32768 waves each,
  //    8 waves/block -> 4096 blocks; blockIdx.y picks Q vs K.
  qk_gemm_wmma<<<dim3(4096, 2), 256, 0, stream>>>(x, q_w, q_b, q_out, k_w, k_b, k_out);

  // 2) Per-edge logits: one wave per edge.
  edge_logits<<<NNZE / 8, 256, 0, stream>>>(q_out, k_out, row_index, col_index,
                                            to_col, att_bias, logits);

  // 3) CSR row pointers from sorted row_index.
  build_row_ptr<<<(NNZE + 1 + 255) / 256, 256, 0, stream>>>(row_index, row_ptr);

  // 4) Fused softmax + weighted segment sums + normalize: one wave per row.
  row_softmax_out<<<NROWS / 8, 256, 0, stream>>>(logits, row_ptr, col_index,
                                                 dist, pos, col_pos, out);
}